// Generator_9466107920813
// MI455X (gfx1250) — compile-verified
//
#include <hip/hip_runtime.h>
#include <hip/hip_bf16.h>

// Fused  out[b, hw, pq] = sigmoid( sum_k w_k * V[b,k,hw] * A[b,k,pq] + bias )
// as a batched GEMM on the f32 WMMA path: C(784x1024) = (w*V)^T(784x16) x A(16x1024)
// K=16 -> exactly 4 steps of V_WMMA_F32_16X16X4_F32 (full precision, no converts).
// Store-bandwidth bound (205 MB output @ 23.3 TB/s ~= 8.8 us floor); NT stores
// keep the >L2-sized output stream from evicting the fully-L2-resident inputs.

typedef __attribute__((ext_vector_type(2))) float v2f;
typedef __attribute__((ext_vector_type(8))) float v8f;

#define BATCH 64
#define KDIM  16
#define MDIM  784   // 28*28 = 49 tiles of 16
#define NDIM  1024  // 32*32 = 64 tiles of 16
#define M_TILES 49
#define N_TILES 64
#define WAVES_PER_BLOCK 8

__global__ __launch_bounds__(256) void bilinear_sigmoid_wmma(
    const float* __restrict__ V,    // (B, K, 784)
    const float* __restrict__ A,    // (B, K, 1024)
    const float* __restrict__ w,    // (K,)
    const float* __restrict__ bptr, // scalar
    float* __restrict__ out)        // (B, 784, 1024)
{
    const int lane = threadIdx.x & 31;
    const int wave = threadIdx.x >> 5;

    // One wave == one full 16x16 output tile (all tiles exact -> EXEC all 1s).
    const int t      = blockIdx.x * WAVES_PER_BLOCK + wave;
    const int n_tile = t & (N_TILES - 1);   // t % 64
    const int rest   = t >> 6;              // t / 64
    const int m_tile = rest % M_TILES;
    const int batch  = rest / M_TILES;

    const int col   = lane & 15;            // M (for A-op) or N (for B-op/D)
    const int khalf = (lane >> 4) << 1;     // lanes 0-15: K pair {k0,k0+1}; 16-31: {k0+2,k0+3}

    const int m = m_tile * 16 + col;
    const int n = n_tile * 16 + col;

    const float* Vb = V + batch * (KDIM * MDIM) + m;   // stride MDIM per k
    const float* Ab = A + batch * (KDIM * NDIM) + n;   // stride NDIM per k

    v8f acc = {};   // C starts at zero

    // K = 16 = 4 x (16x16x4) WMMA steps; fold w into the A (V-side) operand.
#pragma unroll
    for (int s = 0; s < 4; ++s) {
        const int kk = 4 * s + khalf;
        v2f av, bv;
        av.x = w[kk]     * Vb[(size_t)kk * MDIM];
        av.y = w[kk + 1] * Vb[(size_t)(kk + 1) * MDIM];
        bv.x = Ab[(size_t)kk * NDIM];
        bv.y = Ab[(size_t)(kk + 1) * NDIM];
        // (neg_a, A, neg_b, B, c_mod, C, reuse_a, reuse_b)
        acc = __builtin_amdgcn_wmma_f32_16x16x4_f32(
            false, av, false, bv, (short)0, acc, false, false);
    }

    // D layout: VGPR r -> M = r (lanes 0-15) / r+8 (lanes 16-31), N = lane&15.
    const float bias   = *bptr;
    const int   rowoff = (lane >> 4) << 3;  // 0 or 8
    float* orow = out + ((size_t)batch * MDIM + (size_t)(m_tile * 16 + rowoff)) * NDIM + n;

#pragma unroll
    for (int r = 0; r < 8; ++r) {
        const float x = acc[r] + bias;
        // sigmoid via v_exp_f32 + v_rcp_f32 (skip the IEEE div_scale/fixup chain;
        // ~1 ulp rcp is far below activation tolerance)
        const float y = __builtin_amdgcn_rcpf(1.0f + __expf(-x));
        __builtin_nontemporal_store(y, orow + (size_t)r * NDIM);  // 205MB > L2: NT
    }
}

extern "C" void kernel_launch(void* const* d_in, const int* in_sizes, int n_in,
                              void* d_out, int out_size, void* d_ws, size_t ws_size,
                              hipStream_t stream) {
    (void)in_sizes; (void)n_in; (void)out_size; (void)d_ws; (void)ws_size;
    const float* V    = (const float*)d_in[0];
    const float* A    = (const float*)d_in[1];
    const float* w    = (const float*)d_in[2];
    const float* bias = (const float*)d_in[3];
    float* out = (float*)d_out;

    const int total_tiles = BATCH * M_TILES * N_TILES;      // 200704
    const int blocks = total_tiles / WAVES_PER_BLOCK;       // 25088 (exact)
    bilinear_sigmoid_wmma<<<blocks, 256, 0, stream>>>(V, A, w, bias, out);
}